// DeformableAttention3D_14568529068111
// MI455X (gfx1250) — compile-verified
//
#include <hip/hip_runtime.h>
#include <hip/hip_bf16.h>

// Problem constants (from reference setup_inputs)
constexpr int B    = 2;
constexpr int N    = 2048;
constexpr int C    = 128;
constexpr int CAMS = 6;
constexpr int P    = 4;

constexpr int H0 = 32, W0 = 88;
constexpr int H1 = 16, W1 = 44;
constexpr int H2 = 8,  W2 = 22;
constexpr int H3 = 4,  W3 = 11;

typedef float v2f __attribute__((ext_vector_type(2)));
typedef float v8f __attribute__((ext_vector_type(8)));

// ---------------------------------------------------------------------------
// Kernel 1: offsets GEMM (per-point, K=128, 3 outputs) + camera projection.
// One thread per (b, n, p). Writes grids[(b,n,p,cam)][2] and weights[(b,n,p,cam)].
// ---------------------------------------------------------------------------
__global__ void __launch_bounds__(256)
proj_kernel(const float* __restrict__ query,    // [B,N,C]
            const float* __restrict__ means,    // [B,N,3]
            const float* __restrict__ l2i,      // [B,CAMS,4,4]
            const float* __restrict__ Woff,     // [P*3,C]
            const float* __restrict__ boff,     // [P*3]
            const int*   __restrict__ img_h_p,
            const int*   __restrict__ img_w_p,
            float* __restrict__ grids,          // [B,N,P,CAMS,2]
            float* __restrict__ weights)        // [B,N,P,CAMS]
{
    const int idx = blockIdx.x * blockDim.x + threadIdx.x;   // (b*N+n)*P + p
    if (idx >= B * N * P) return;
    const int p  = idx % P;
    const int bn = idx / P;          // b*N + n
    const int b  = bn / N;

    const float* q  = query + (size_t)bn * C;
    const float* w0 = Woff + (size_t)(p * 3 + 0) * C;
    const float* w1 = Woff + (size_t)(p * 3 + 1) * C;
    const float* w2 = Woff + (size_t)(p * 3 + 2) * C;

    float o0 = boff[p * 3 + 0];
    float o1 = boff[p * 3 + 1];
    float o2 = boff[p * 3 + 2];
#pragma unroll 8
    for (int k = 0; k < C; ++k) {
        const float qv = q[k];
        o0 = fmaf(qv, w0[k], o0);
        o1 = fmaf(qv, w1[k], o1);
        o2 = fmaf(qv, w2[k], o2);
    }

    const float rx = means[(size_t)bn * 3 + 0] + o0;
    const float ry = means[(size_t)bn * 3 + 1] + o1;
    const float rz = means[(size_t)bn * 3 + 2] + o2;

    const float fw = (float)img_w_p[0];
    const float fh = (float)img_h_p[0];

    float gxv[CAMS], gyv[CAMS];
    int   vfl[CAMS];
    int   vcnt = 0;
#pragma unroll
    for (int cam = 0; cam < CAMS; ++cam) {
        const float* M = l2i + (size_t)(b * CAMS + cam) * 16;
        const float px = M[0] * rx + M[1] * ry + M[2]  * rz + M[3];
        const float py = M[4] * rx + M[5] * ry + M[6]  * rz + M[7];
        const float pz = M[8] * rx + M[9] * ry + M[10] * rz + M[11];
        const float d  = fmaxf(pz, 0.001f);
        const float gx = 2.0f * (px / d) / fw - 1.0f;
        const float gy = 2.0f * (py / d) / fh - 1.0f;
        const int v = (fabsf(gx) <= 1.0f) && (fabsf(gy) <= 1.0f);
        gxv[cam] = gx; gyv[cam] = gy; vfl[cam] = v; vcnt += v;
    }
    const float inv = 1.0f / fmaxf((float)vcnt, 1.0f);

    const size_t base = (size_t)idx * CAMS;
#pragma unroll
    for (int cam = 0; cam < CAMS; ++cam) {
        grids[(base + cam) * 2 + 0] = gxv[cam];
        grids[(base + cam) * 2 + 1] = gyv[cam];
        weights[base + cam] = vfl[cam] ? inv : 0.0f;
    }
}

// ---------------------------------------------------------------------------
// Transpose one pyramid level [BC, C, HW] -> [BC, HW, C] (channels-last).
// Coalesced reads; one-time ~23 MB total across the 4 levels.
// ---------------------------------------------------------------------------
__global__ void __launch_bounds__(256)
transpose_kernel(const float* __restrict__ in, float* __restrict__ out, int HW)
{
    const int total = B * CAMS * C * HW;
    int idx = blockIdx.x * blockDim.x + threadIdx.x;
    if (idx >= total) return;
    const int hw  = idx % HW;
    const int c   = (idx / HW) % C;
    const int bc  = idx / (HW * C);
    out[((size_t)bc * HW + hw) * C + c] = in[idx];
}

// ---------------------------------------------------------------------------
// Kernel 2 (fast path): bilinear grid-sample from channels-last features.
// One block per (b,n); 128 threads = channels. Each corner fetch per wave is
// 32 consecutive floats (128 B) -> fully coalesced, minimal L2 line traffic.
// ---------------------------------------------------------------------------
__global__ void __launch_bounds__(128)
sample_kernel_t(const float* __restrict__ t0, const float* __restrict__ t1,
                const float* __restrict__ t2, const float* __restrict__ t3,
                const float* __restrict__ grids,    // [B,N,P,CAMS,2]
                const float* __restrict__ weights,  // [B,N,P,CAMS]
                float* __restrict__ agg)            // [B,N,C]
{
    const int bn = blockIdx.x;       // b*N + n
    const int c  = threadIdx.x;      // channel
    const int b  = bn / N;

    __shared__ float sg[P * CAMS * 2];
    __shared__ float sw[P * CAMS];
    const size_t gbase = (size_t)bn * P * CAMS;
    if (c < P * CAMS * 2)      sg[c] = grids[gbase * 2 + c];
    else if (c < P * CAMS * 3) sw[c - P * CAMS * 2] = weights[gbase + (c - P * CAMS * 2)];
    __syncthreads();

    const float* fps[4] = {t0, t1, t2, t3};
    const int Hs[4] = {H0, H1, H2, H3};
    const int Ws[4] = {W0, W1, W2, W3};

    float acc = 0.0f;
    for (int pc = 0; pc < P * CAMS; ++pc) {
        const float w = sw[pc];
        if (w == 0.0f) continue;              // uniform across block
        const int cam = pc % CAMS;
        const int bc  = b * CAMS + cam;
        const float gx = sg[pc * 2 + 0];
        const float gy = sg[pc * 2 + 1];
#pragma unroll
        for (int l = 0; l < 4; ++l) {
            const int H = Hs[l], W = Ws[l];
            const float x = (gx + 1.0f) * 0.5f * (float)W - 0.5f;
            const float y = (gy + 1.0f) * 0.5f * (float)H - 0.5f;
            const float xf = floorf(x), yf = floorf(y);
            const float wx = x - xf, wy = y - yf;
            const int x0 = (int)xf, y0 = (int)yf;
            const int x1 = x0 + 1,  y1 = y0 + 1;
            // channels-last: element (bc, y, x, c)
            const float* fb = fps[l] + ((size_t)bc * H * W) * C + c;
            const bool vx0 = (x0 >= 0) & (x0 < W);
            const bool vx1 = (x1 >= 0) & (x1 < W);
            const bool vy0 = (y0 >= 0) & (y0 < H);
            const bool vy1 = (y1 >= 0) & (y1 < H);
            const float v00 = (vx0 & vy0) ? fb[(size_t)(y0 * W + x0) * C] : 0.0f;
            const float v01 = (vx1 & vy0) ? fb[(size_t)(y0 * W + x1) * C] : 0.0f;
            const float v10 = (vx0 & vy1) ? fb[(size_t)(y1 * W + x0) * C] : 0.0f;
            const float v11 = (vx1 & vy1) ? fb[(size_t)(y1 * W + x1) * C] : 0.0f;
            const float val = v00 * (1.0f - wx) * (1.0f - wy)
                            + v01 * wx          * (1.0f - wy)
                            + v10 * (1.0f - wx) * wy
                            + v11 * wx          * wy;
            acc = fmaf(w, val, acc);
        }
    }
    agg[(size_t)bn * C + c] = acc * (1.0f / 16.0f);   // mean over 4 levels * P
}

// ---------------------------------------------------------------------------
// Kernel 2 (fallback): sample directly from original [BC, C, H, W] layout.
// Used only if the workspace is too small for the channels-last copies.
// ---------------------------------------------------------------------------
__global__ void __launch_bounds__(128)
sample_kernel(const float* __restrict__ f0, const float* __restrict__ f1,
              const float* __restrict__ f2, const float* __restrict__ f3,
              const float* __restrict__ grids,
              const float* __restrict__ weights,
              float* __restrict__ agg)
{
    const int bn = blockIdx.x;
    const int c  = threadIdx.x;
    const int b  = bn / N;

    __shared__ float sg[P * CAMS * 2];
    __shared__ float sw[P * CAMS];
    const size_t gbase = (size_t)bn * P * CAMS;
    if (c < P * CAMS * 2)      sg[c] = grids[gbase * 2 + c];
    else if (c < P * CAMS * 3) sw[c - P * CAMS * 2] = weights[gbase + (c - P * CAMS * 2)];
    __syncthreads();

    const float* fps[4] = {f0, f1, f2, f3};
    const int Hs[4] = {H0, H1, H2, H3};
    const int Ws[4] = {W0, W1, W2, W3};

    float acc = 0.0f;
    for (int pc = 0; pc < P * CAMS; ++pc) {
        const float w = sw[pc];
        if (w == 0.0f) continue;
        const int cam = pc % CAMS;
        const float gx = sg[pc * 2 + 0];
        const float gy = sg[pc * 2 + 1];
#pragma unroll
        for (int l = 0; l < 4; ++l) {
            const int H = Hs[l], W = Ws[l];
            const float x = (gx + 1.0f) * 0.5f * (float)W - 0.5f;
            const float y = (gy + 1.0f) * 0.5f * (float)H - 0.5f;
            const float xf = floorf(x), yf = floorf(y);
            const float wx = x - xf, wy = y - yf;
            const int x0 = (int)xf, y0 = (int)yf;
            const int x1 = x0 + 1,  y1 = y0 + 1;
            const float* fb = fps[l] + ((size_t)(b * CAMS + cam) * C + c) * (size_t)(H * W);
            const bool vx0 = (x0 >= 0) & (x0 < W);
            const bool vx1 = (x1 >= 0) & (x1 < W);
            const bool vy0 = (y0 >= 0) & (y0 < H);
            const bool vy1 = (y1 >= 0) & (y1 < H);
            const float v00 = (vx0 & vy0) ? fb[y0 * W + x0] : 0.0f;
            const float v01 = (vx1 & vy0) ? fb[y0 * W + x1] : 0.0f;
            const float v10 = (vx0 & vy1) ? fb[y1 * W + x0] : 0.0f;
            const float v11 = (vx1 & vy1) ? fb[y1 * W + x1] : 0.0f;
            const float val = v00 * (1.0f - wx) * (1.0f - wy)
                            + v01 * wx          * (1.0f - wy)
                            + v10 * (1.0f - wx) * wy
                            + v11 * wx          * wy;
            acc = fmaf(w, val, acc);
        }
    }
    agg[(size_t)bn * C + c] = acc * (1.0f / 16.0f);
}

// ---------------------------------------------------------------------------
// Kernel 3: out = agg @ W_out^T + b_out  via V_WMMA_F32_16X16X4_F32.
// M = B*N = 4096, N = 128, K = 128.  One 16x16 tile per wave, 32 WMMA / tile.
// Fragment layouts per ISA 7.12.2:
//   A (16x4 f32):  lane L -> row L%16, K-half L/16, 2 K-values per lane (v2f)
//   B (4x16 f32):  lane L -> col L%16, K-half L/16, 2 K-values per lane (v2f)
//   C/D (16x16):   lane L, reg r -> (M = r + 8*(L/16), N = L%16)
// ---------------------------------------------------------------------------
__global__ void __launch_bounds__(256)
out_gemm_kernel(const float* __restrict__ Amat,   // agg  [4096][128]
                const float* __restrict__ Wout,   // W_out[128][128] (out = A * Wout^T)
                const float* __restrict__ bias,   // [128]
                float* __restrict__ out)          // [4096][128]
{
    const int lane = threadIdx.x & 31;
    const int wave = threadIdx.x >> 5;
    const int tile = blockIdx.x * 8 + wave;       // 2048 tiles total
    const int NTN  = C / 16;                      // 8
    const int m0   = (tile / NTN) * 16;
    const int n0   = (tile % NTN) * 16;
    const int lm   = lane & 15;                   // row (A) / col (B,D)
    const int lh   = lane >> 4;                   // K-half selector

    const float* arow = Amat + (size_t)(m0 + lm) * C + 2 * lh;
    const float* brow = Wout + (size_t)(n0 + lm) * C + 2 * lh;   // B(k,n) = Wout[n][k]

    v8f acc = {0.f, 0.f, 0.f, 0.f, 0.f, 0.f, 0.f, 0.f};
#pragma unroll
    for (int kk = 0; kk < C; kk += 4) {
        v2f a = *reinterpret_cast<const v2f*>(arow + kk);
        v2f b = *reinterpret_cast<const v2f*>(brow + kk);
        acc = __builtin_amdgcn_wmma_f32_16x16x4_f32(
            /*neg_a=*/false, a, /*neg_b=*/false, b,
            /*c_mod=*/(short)0, acc, /*reuse_a=*/false, /*reuse_b=*/false);
    }

    const float bv = bias[n0 + lm];
#pragma unroll
    for (int r = 0; r < 8; ++r) {
        const int m = m0 + r + 8 * lh;
        out[(size_t)m * C + (n0 + lm)] = acc[r] + bv;
    }
}

// ---------------------------------------------------------------------------
// Host launcher
// ---------------------------------------------------------------------------
extern "C" void kernel_launch(void* const* d_in, const int* in_sizes, int n_in,
                              void* d_out, int out_size, void* d_ws, size_t ws_size,
                              hipStream_t stream) {
    const float* query  = (const float*)d_in[0];
    const float* means  = (const float*)d_in[1];
    const float* feat0  = (const float*)d_in[2];
    const float* feat1  = (const float*)d_in[3];
    const float* feat2  = (const float*)d_in[4];
    const float* feat3  = (const float*)d_in[5];
    // d_in[6] = depth_maps (unused by the reference computation)
    const float* l2i    = (const float*)d_in[7];
    const float* Woff   = (const float*)d_in[8];
    const float* boff   = (const float*)d_in[9];
    const float* Wout   = (const float*)d_in[10];
    const float* bout   = (const float*)d_in[11];
    const int*   img_h  = (const int*)d_in[12];
    const int*   img_w  = (const int*)d_in[13];
    float* out = (float*)d_out;

    // Workspace layout (all float32)
    const size_t n_grids  = (size_t)B * N * P * CAMS * 2;
    const size_t n_weight = (size_t)B * N * P * CAMS;
    const size_t n_agg    = (size_t)B * N * C;
    const int    HWs[4]   = {H0 * W0, H1 * W1, H2 * W2, H3 * W3};
    size_t n_tfeat = 0;
    for (int l = 0; l < 4; ++l) n_tfeat += (size_t)B * CAMS * C * HWs[l];

    float* grids   = (float*)d_ws;
    float* weightw = grids + n_grids;
    float* agg     = weightw + n_weight;
    float* tbase   = agg + n_agg;

    const size_t need_bytes = (n_grids + n_weight + n_agg + n_tfeat) * sizeof(float);
    const bool use_transposed = (ws_size >= need_bytes);

    {   // Phase A: offsets + projection
        const int total = B * N * P;                              // 16384
        proj_kernel<<<(total + 255) / 256, 256, 0, stream>>>(
            query, means, l2i, Woff, boff, img_h, img_w, grids, weightw);
    }

    if (use_transposed) {
        // Phase A': one-time channels-last transposes of the feature pyramid
        const float* fin[4] = {feat0, feat1, feat2, feat3};
        float* tf[4];
        size_t off = 0;
        for (int l = 0; l < 4; ++l) { tf[l] = tbase + off; off += (size_t)B * CAMS * C * HWs[l]; }
        for (int l = 0; l < 4; ++l) {
            const int total = B * CAMS * C * HWs[l];
            transpose_kernel<<<(total + 255) / 256, 256, 0, stream>>>(fin[l], tf[l], HWs[l]);
        }
        // Phase B: coalesced grid sample + aggregation
        sample_kernel_t<<<B * N, C, 0, stream>>>(
            tf[0], tf[1], tf[2], tf[3], grids, weightw, agg);
    } else {
        // Phase B (fallback): strided gather from original layout
        sample_kernel<<<B * N, C, 0, stream>>>(
            feat0, feat1, feat2, feat3, grids, weightw, agg);
    }

    {   // Phase C: output projection (WMMA fp32)
        const int tiles = (B * N / 16) * (C / 16);                // 2048
        out_gemm_kernel<<<tiles / 8, 256, 0, stream>>>(agg, Wout, bout, out);
    }
    (void)in_sizes; (void)n_in; (void)out_size;
}